// VariationalGCNEncoder_43722767073850
// MI455X (gfx1250) — compile-verified
//
#include <hip/hip_runtime.h>
#include <hip/hip_bf16.h>
#include <stdint.h>

typedef __attribute__((ext_vector_type(2))) float v2f;
typedef __attribute__((ext_vector_type(8))) float v8f;

#define IN_C  128
#define HID   256
#define OUT_C 64
#define LDS_PAD 4   // stride % 64 banks == 4 -> conflict-free 16-lane column reads

// async global->LDS copy of 16 bytes (CDNA5 GLOBAL_LOAD_ASYNC_TO_LDS_B128, ASYNCcnt)
__device__ __forceinline__ void async_copy_b128(unsigned lds_off, const float* gsrc) {
  asm volatile("global_load_async_to_lds_b128 %0, %1, off"
               :: "v"(lds_off), "v"((unsigned long long)(uintptr_t)gsrc)
               : "memory");
}
__device__ __forceinline__ void async_wait0() {
  asm volatile("s_wait_asynccnt 0x0" ::: "memory");
}

// ---------------- degree / normalization ----------------

__global__ void init_deg_kernel(float* __restrict__ deg, int n) {
  int i = blockIdx.x * blockDim.x + threadIdx.x;
  if (i < n) deg[i] = 1.0f;  // self loop contributes 1
}

__global__ void deg_scatter_kernel(const long long* __restrict__ dst,
                                   float* __restrict__ deg, int e) {
  int i = blockIdx.x * blockDim.x + threadIdx.x;
  if (i < e) atomicAdd(&deg[(int)dst[i]], 1.0f);
}

__global__ void dinv_kernel(float* __restrict__ deg, int n) {
  int i = blockIdx.x * blockDim.x + threadIdx.x;
  if (i < n) deg[i] = rsqrtf(deg[i]);  // deg >= 1 always (self loop)
}

// ---------------- layer 1: aggregate-first (A @ x), 128 channels ----------------

__global__ void agg_init_kernel(const float* __restrict__ x,
                                const float* __restrict__ dinv,
                                float* __restrict__ agg, int n) {
  int i = blockIdx.x * blockDim.x + threadIdx.x;
  if (i < n * IN_C) {
    int node = i >> 7;  // /128
    float di = dinv[node];
    agg[i] = di * di * x[i];  // self-loop term
  }
}

// one wave32 per edge; each lane moves 4 channels (float4)
__global__ void agg_edges_kernel(const float* __restrict__ x,
                                 const long long* __restrict__ src,
                                 const long long* __restrict__ dst,
                                 const float* __restrict__ dinv,
                                 float* __restrict__ agg, int e) {
  int wave = (int)((blockIdx.x * blockDim.x + threadIdx.x) >> 5);
  int lane = threadIdx.x & 31;
  if (wave >= e) return;
  int s = (int)src[wave];
  int d = (int)dst[wave];
  float w = dinv[s] * dinv[d];
  const float4* xr = (const float4*)(x + (size_t)s * IN_C);
  float4 v = xr[lane];
  float* ar = agg + (size_t)d * IN_C + lane * 4;
  atomicAdd(ar + 0, w * v.x);
  atomicAdd(ar + 1, w * v.y);
  atomicAdd(ar + 2, w * v.z);
  atomicAdd(ar + 3, w * v.w);
}

// ---------------- GEMM1: h = relu(agg @ W1 + b1), WMMA f32 16x16x4 ----------------
// A tile (16x128) staged once per block in LDS via async copy, shared by 8 waves.
// A layout (32-bit 16x4): lanes 0-15 hold (M=lane, K={0,1}); lanes 16-31 K={2,3}
// D layout: VGPR j -> (M = j + 8*half, N = lane%16)

__global__ __launch_bounds__(256) void gemm1_kernel(const float* __restrict__ A,
                                                    const float* __restrict__ W,
                                                    const float* __restrict__ bias,
                                                    float* __restrict__ H, int n) {
  __shared__ float tileA[16 * (IN_C + LDS_PAD)];
  const int tid   = threadIdx.x;
  const int mtile = blockIdx.x;

  // stage 16x128 A tile: 512 float4s, 2 per thread, async global->LDS
  for (int i = tid; i < 512; i += 256) {
    int r  = i >> 5;       // float4-row (32 float4 per row)
    int c4 = i & 31;
    int grow = mtile * 16 + r;
    if (grow >= n) grow = n - 1;
    const float* gsrc = A + (size_t)grow * IN_C + c4 * 4;
    unsigned loff = (unsigned)(uintptr_t)&tileA[r * (IN_C + LDS_PAD) + c4 * 4];
    async_copy_b128(loff, gsrc);
  }
  async_wait0();
  __syncthreads();

  const int wave = tid >> 5;
  const int lane = tid & 31;
  const int half = lane >> 4;
  const int ml   = lane & 15;
  const int ntile = blockIdx.y * 8 + wave;  // 16 n-tiles over HID=256
  const int col   = ntile * 16 + ml;

  const float* arow = &tileA[ml * (IN_C + LDS_PAD) + 2 * half];  // LDS, conflict-free
  const float* bcol = W + (size_t)(2 * half) * HID + col;

  v8f acc = {};
  for (int k = 0; k < IN_C; k += 4) {
    v2f a = *(const v2f*)(arow + k);                 // ds_load_b64
    v2f b;
    b.x = bcol[(size_t)k * HID];                     // W[k+2h][col]
    b.y = bcol[(size_t)(k + 1) * HID];               // W[k+2h+1][col]
    acc = __builtin_amdgcn_wmma_f32_16x16x4_f32(false, a, false, b,
                                                (short)0, acc, false, false);
  }

  float bv = bias[col];
  int obase = mtile * 16 + 8 * half;
  float* out = H + (size_t)obase * HID + ntile * 16 + ml;
  for (int j = 0; j < 8; ++j) {
    int r = obase + j;
    if (r < n) out[(size_t)j * HID] = fmaxf(acc[j] + bv, 0.0f);
  }
}

// ---------------- GEMMs 2&3 fused: hm = h @ W_mu, hl = h @ W_ls ----------------

__global__ __launch_bounds__(256) void gemm23_kernel(const float* __restrict__ H,
                                                     const float* __restrict__ Wmu,
                                                     const float* __restrict__ Wls,
                                                     float* __restrict__ HM,
                                                     float* __restrict__ HL, int n) {
  __shared__ float tileA[16 * (HID + LDS_PAD)];
  const int tid   = threadIdx.x;
  const int mtile = blockIdx.x;

  // stage 16x256 A tile: 1024 float4s, 4 per thread, async global->LDS
  for (int i = tid; i < 1024; i += 256) {
    int r  = i >> 6;       // 64 float4 per row
    int c4 = i & 63;
    int grow = mtile * 16 + r;
    if (grow >= n) grow = n - 1;
    const float* gsrc = H + (size_t)grow * HID + c4 * 4;
    unsigned loff = (unsigned)(uintptr_t)&tileA[r * (HID + LDS_PAD) + c4 * 4];
    async_copy_b128(loff, gsrc);
  }
  async_wait0();
  __syncthreads();

  const int wave = tid >> 5;
  const int lane = tid & 31;
  const int half = lane >> 4;
  const int ml   = lane & 15;

  const float* W = (wave < 4) ? Wmu : Wls;
  float*       O = (wave < 4) ? HM  : HL;
  const int ntile = wave & 3;  // 4 n-tiles over OUT_C=64
  const int col   = ntile * 16 + ml;

  const float* arow = &tileA[ml * (HID + LDS_PAD) + 2 * half];
  const float* bcol = W + (size_t)(2 * half) * OUT_C + col;

  v8f acc = {};
  for (int k = 0; k < HID; k += 4) {
    v2f a = *(const v2f*)(arow + k);
    v2f b;
    b.x = bcol[(size_t)k * OUT_C];
    b.y = bcol[(size_t)(k + 1) * OUT_C];
    acc = __builtin_amdgcn_wmma_f32_16x16x4_f32(false, a, false, b,
                                                (short)0, acc, false, false);
  }

  int obase = mtile * 16 + 8 * half;
  float* out = O + (size_t)obase * OUT_C + ntile * 16 + ml;
  for (int j = 0; j < 8; ++j) {
    int r = obase + j;
    if (r < n) out[(size_t)j * OUT_C] = acc[j];
  }
}

// ---------------- output: self-loop + bias init, then edge scatter ----------------

__global__ void out_init_kernel(const float* __restrict__ hm, const float* __restrict__ hl,
                                const float* __restrict__ dinv,
                                const float* __restrict__ bmu, const float* __restrict__ bls,
                                float* __restrict__ mu, float* __restrict__ ls, int n) {
  int i = blockIdx.x * blockDim.x + threadIdx.x;
  if (i < n * OUT_C) {
    int node = i >> 6;  // /64
    int c = i & 63;
    float di = dinv[node];
    float w = di * di;
    mu[i] = w * hm[i] + bmu[c];
    ls[i] = w * hl[i] + bls[c];
  }
}

// one wave32 per edge; lane handles 2 channels of mu and 2 of logstd
__global__ void out_edges_kernel(const float* __restrict__ hm, const float* __restrict__ hl,
                                 const long long* __restrict__ src,
                                 const long long* __restrict__ dst,
                                 const float* __restrict__ dinv,
                                 float* __restrict__ mu, float* __restrict__ ls, int e) {
  int wave = (int)((blockIdx.x * blockDim.x + threadIdx.x) >> 5);
  int lane = threadIdx.x & 31;
  if (wave >= e) return;
  int s = (int)src[wave];
  int d = (int)dst[wave];
  float w = dinv[s] * dinv[d];
  const float2* ms  = (const float2*)(hm + (size_t)s * OUT_C);
  const float2* lsv = (const float2*)(hl + (size_t)s * OUT_C);
  float2 mv = ms[lane];
  float2 lv = lsv[lane];
  float* mo = mu + (size_t)d * OUT_C + lane * 2;
  float* lo = ls + (size_t)d * OUT_C + lane * 2;
  atomicAdd(mo + 0, w * mv.x);
  atomicAdd(mo + 1, w * mv.y);
  atomicAdd(lo + 0, w * lv.x);
  atomicAdd(lo + 1, w * lv.y);
}

// ---------------- launch ----------------

extern "C" void kernel_launch(void* const* d_in, const int* in_sizes, int n_in,
                              void* d_out, int out_size, void* d_ws, size_t ws_size,
                              hipStream_t stream) {
  const float*     x    = (const float*)d_in[0];
  const long long* ei   = (const long long*)d_in[1];
  const float*     W1   = (const float*)d_in[2];
  const float*     b1   = (const float*)d_in[3];
  const float*     Wmu  = (const float*)d_in[4];
  const float*     bmu  = (const float*)d_in[5];
  const float*     Wls  = (const float*)d_in[6];
  const float*     bls  = (const float*)d_in[7];

  int n = in_sizes[0] / IN_C;        // 100000 nodes
  int e = in_sizes[1] / 2;           // 1.6M edges
  const long long* src = ei;
  const long long* dst = ei + e;

  float* mu = (float*)d_out;
  float* ls = (float*)d_out + (size_t)n * OUT_C;

  // workspace layout (floats): dinv[n] | agg[n*128] (reused as hm[n*64]+hl[n*64]) | h[n*256]
  float* dinv = (float*)d_ws;
  float* agg  = dinv + n;
  float* h    = agg + (size_t)n * IN_C;
  float* hmv  = agg;                        // reuse agg space after GEMM1
  float* hlv  = agg + (size_t)n * OUT_C;

  int mtiles = (n + 15) / 16;

  // 1) symmetric normalization
  init_deg_kernel<<<(n + 255) / 256, 256, 0, stream>>>(dinv, n);
  deg_scatter_kernel<<<(e + 255) / 256, 256, 0, stream>>>(dst, dinv, e);
  dinv_kernel<<<(n + 255) / 256, 256, 0, stream>>>(dinv, n);

  // 2) layer 1 aggregation (aggregate-first: 128 channels)
  agg_init_kernel<<<((size_t)n * IN_C + 255) / 256, 256, 0, stream>>>(x, dinv, agg, n);
  agg_edges_kernel<<<(e + 7) / 8, 256, 0, stream>>>(x, src, dst, dinv, agg, e);

  // 3) GEMM1 + bias + relu -> h
  gemm1_kernel<<<dim3(mtiles, 2), 256, 0, stream>>>(agg, W1, b1, h, n);

  // 4) fused GEMMs for mu / logstd projections (transform-first: 64 channels each)
  gemm23_kernel<<<dim3(mtiles, 1), 256, 0, stream>>>(h, Wmu, Wls, hmv, hlv, n);

  // 5) output aggregation
  out_init_kernel<<<((size_t)n * OUT_C + 255) / 256, 256, 0, stream>>>(
      hmv, hlv, dinv, bmu, bls, mu, ls, n);
  out_edges_kernel<<<(e + 7) / 8, 256, 0, stream>>>(hmv, hlv, src, dst, dinv, mu, ls, e);
}